// RegionalEncoder_52390011077090
// MI455X (gfx1250) — compile-verified
//
#include <hip/hip_runtime.h>
#include <hip/hip_bf16.h>
#include <math.h>

// ---------------------------------------------------------------------------
// 2-layer GAT for MI455X (gfx1250, wave32).
//  - Dense GEMMs: V_WMMA_F32_16X16X4_F32, one wave per 16x64 output strip
//    (4 accumulators) so each A float2 load feeds 4 WMMAs.
//  - Edge softmax/scatter: HBM-roofline bound (~1 GB traffic -> ~50us @23.3TB/s).
//    int64 edge_index packed once to int2; float4-vectorized streaming passes
//    with f32 atomics (global_atomic_add_f32 / max_i32 / min_u32).
// ---------------------------------------------------------------------------

#define N_NODES   100000
#define N_EDGES   1600000
#define E_TOT     (N_EDGES + N_NODES)   // with self loops
#define HEADS     4
#define CH        16                    // per-head channels (64/4, both layers)
#define DIM       64
#define NEG_SLOPE 0.2f

typedef __attribute__((ext_vector_type(2))) float v2f;
typedef __attribute__((ext_vector_type(8))) float v8f;

// ---------------- WMMA f32 GEMM: C[M,64] = A[M,K] @ B[K,64] (+bias) ---------
// One wave per 16-row strip across all 4 N-tiles. M%16==0, K%4==0.
// A-layout (ISA 7.12.2, 32-bit A 16x4): lanes 0-15 -> M=lane, K={0,1};
// lanes 16-31 -> M=lane-16, K={2,3}. B/C/D: N striped across lanes.
__global__ void gemm_wmma_strip(const float* __restrict__ A,
                                const float* __restrict__ B,
                                const float* __restrict__ bias,
                                float* __restrict__ C,
                                int M, int K) {
  const int wid  = (blockIdx.x * blockDim.x + threadIdx.x) >> 5;
  const int lane = threadIdx.x & 31;
  if (wid >= (M >> 4)) return;              // wave-uniform: EXEC stays all-1s
  const int half = lane >> 4;               // 0 or 1
  const int l15  = lane & 15;
  const float* Ap = A + (size_t)(wid * 16 + l15) * K + 2 * half;
  const float* Bp = B + l15;

  v8f acc0 = {}, acc1 = {}, acc2 = {}, acc3 = {};
  for (int k = 0; k < K; k += 4) {
    const v2f a = *(const v2f*)(Ap + k);             // global_load_b64
    const float* Br0 = Bp + (size_t)(k + 2 * half) * DIM;
    const float* Br1 = Br0 + DIM;
    v2f b0, b1, b2, b3;
    b0.x = Br0[0];  b0.y = Br1[0];
    b1.x = Br0[16]; b1.y = Br1[16];
    b2.x = Br0[32]; b2.y = Br1[32];
    b3.x = Br0[48]; b3.y = Br1[48];
    acc0 = __builtin_amdgcn_wmma_f32_16x16x4_f32(false, a, false, b0, (short)0, acc0, false, false);
    acc1 = __builtin_amdgcn_wmma_f32_16x16x4_f32(false, a, false, b1, (short)0, acc1, false, false);
    acc2 = __builtin_amdgcn_wmma_f32_16x16x4_f32(false, a, false, b2, (short)0, acc2, false, false);
    acc3 = __builtin_amdgcn_wmma_f32_16x16x4_f32(false, a, false, b3, (short)0, acc3, false, false);
  }

  const int mbase = wid * 16 + 8 * half;    // lanes 16-31 hold M=8..15
  v8f accs[4] = {acc0, acc1, acc2, acc3};
#pragma unroll
  for (int nt = 0; nt < 4; ++nt) {
    const int col = nt * 16 + l15;
    const float bv = bias ? bias[col] : 0.0f;
#pragma unroll
    for (int r = 0; r < 8; ++r)
      C[(size_t)(mbase + r) * DIM + col] = accs[nt][r] + bv;
  }
}

// ---------------- pack int64 edge_index (+self loops) into int2 -------------
__global__ void build_edges(const long long* __restrict__ ei,
                            int2* __restrict__ sd) {
  const int e = blockIdx.x * blockDim.x + threadIdx.x;
  if (e >= E_TOT) return;
  int s, d;
  if (e < N_EDGES) {
    s = (int)ei[e];
    d = (int)ei[(size_t)N_EDGES + e];
  } else {
    s = d = e - N_EDGES;                    // self loop
  }
  sd[e] = make_int2(s, d);
}

// ---------------- per-(node,head) attention logits --------------------------
__global__ void att_logits(const float* __restrict__ xw,
                           const float* __restrict__ a_src,
                           const float* __restrict__ a_dst,
                           float* __restrict__ al_src,
                           float* __restrict__ al_dst) {
  const int i = blockIdx.x * blockDim.x + threadIdx.x;   // node*HEADS + h
  if (i >= N_NODES * HEADS) return;
  const int h = i & (HEADS - 1);
  const float* v = xw + (size_t)(i >> 2) * DIM + h * CH;
  float ss = 0.f, sd = 0.f;
#pragma unroll
  for (int c = 0; c < CH; ++c) {
    const float x = v[c];
    ss += x * a_src[h * CH + c];
    sd += x * a_dst[h * CH + c];
  }
  al_src[i] = ss;
  al_dst[i] = sd;
}

// ---------------- per-layer state init (deterministic each call) ------------
__global__ void init_layer(float4* __restrict__ m, float4* __restrict__ s,
                           float4* __restrict__ agg) {
  const int i = blockIdx.x * blockDim.x + threadIdx.x;   // float4 index
  if (i < N_NODES * (DIM / 4)) agg[i] = make_float4(0.f, 0.f, 0.f, 0.f);
  if (i < N_NODES) {
    m[i] = make_float4(-INFINITY, -INFINITY, -INFINITY, -INFINITY);
    s[i] = make_float4(0.f, 0.f, 0.f, 0.f);
  }
}

// float atomic-max via int-order trick (no CAS loop):
// nonneg floats order like ints; negative floats order reversed as unsigned.
__device__ __forceinline__ void atomicMaxF(float* addr, float v) {
  if (v >= 0.f) atomicMax((int*)addr, __float_as_int(v));
  else          atomicMin((unsigned int*)addr, (unsigned int)__float_as_int(v));
}

__device__ __forceinline__ float lrelu(float v) {
  return v > 0.f ? v : NEG_SLOPE * v;
}

// ---------------- pass 1: leaky-relu logits + segment max -------------------
__global__ void edge_logits_max(const int2* __restrict__ sd,
                                const float4* __restrict__ al_src,
                                const float4* __restrict__ al_dst,
                                float4* __restrict__ ebuf,
                                float* __restrict__ m) {
  const int e = blockIdx.x * blockDim.x + threadIdx.x;
  if (e >= E_TOT) return;
  const int2 p = sd[e];
  const float4 as = al_src[p.x];            // global_load_b128
  const float4 ad = al_dst[p.y];
  float4 v;
  v.x = lrelu(as.x + ad.x);
  v.y = lrelu(as.y + ad.y);
  v.z = lrelu(as.z + ad.z);
  v.w = lrelu(as.w + ad.w);
  ebuf[e] = v;
  float* mp = m + (size_t)p.y * HEADS;
  atomicMaxF(mp + 0, v.x);
  atomicMaxF(mp + 1, v.y);
  atomicMaxF(mp + 2, v.z);
  atomicMaxF(mp + 3, v.w);
}

// ---------------- pass 2: exp(e - max) + segment sum ------------------------
__global__ void edge_expsum(const int2* __restrict__ sd,
                            const float4* __restrict__ m,
                            float4* __restrict__ ebuf,
                            float* __restrict__ ssum) {
  const int e = blockIdx.x * blockDim.x + threadIdx.x;
  if (e >= E_TOT) return;
  const int2 p = sd[e];
  float4 mm = m[p.y];
  if (!isfinite(mm.x)) mm.x = 0.f;
  if (!isfinite(mm.y)) mm.y = 0.f;
  if (!isfinite(mm.z)) mm.z = 0.f;
  if (!isfinite(mm.w)) mm.w = 0.f;
  float4 v = ebuf[e];
  v.x = expf(v.x - mm.x);
  v.y = expf(v.y - mm.y);
  v.z = expf(v.z - mm.z);
  v.w = expf(v.w - mm.w);
  ebuf[e] = v;
  float* sp = ssum + (size_t)p.y * HEADS;
  atomicAdd(sp + 0, v.x);
  atomicAdd(sp + 1, v.y);
  atomicAdd(sp + 2, v.z);
  atomicAdd(sp + 3, v.w);
}

// ---------------- pass 3: alpha-weighted message scatter --------------------
// One thread per (edge, head); float4 gathers + 16 f32 atomic adds.
__global__ void edge_scatter(const int2* __restrict__ sd,
                             const float* __restrict__ ebuf,
                             const float* __restrict__ ssum,
                             const float* __restrict__ xw,
                             float* __restrict__ agg) {
  const int t = blockIdx.x * blockDim.x + threadIdx.x;
  if (t >= E_TOT * HEADS) return;
  const int e = t >> 2;
  const int h = t & 3;
  const int2 p = sd[e];
  const float alpha = ebuf[t] / (ssum[p.y * HEADS + h] + 1e-16f);
  const float4* sp = (const float4*)(xw + (size_t)p.x * DIM + h * CH);
  float*        dp = agg + (size_t)p.y * DIM + h * CH;
  const float4 m0 = sp[0], m1 = sp[1], m2 = sp[2], m3 = sp[3];
  atomicAdd(dp + 0,  m0.x * alpha);
  atomicAdd(dp + 1,  m0.y * alpha);
  atomicAdd(dp + 2,  m0.z * alpha);
  atomicAdd(dp + 3,  m0.w * alpha);
  atomicAdd(dp + 4,  m1.x * alpha);
  atomicAdd(dp + 5,  m1.y * alpha);
  atomicAdd(dp + 6,  m1.z * alpha);
  atomicAdd(dp + 7,  m1.w * alpha);
  atomicAdd(dp + 8,  m2.x * alpha);
  atomicAdd(dp + 9,  m2.y * alpha);
  atomicAdd(dp + 10, m2.z * alpha);
  atomicAdd(dp + 11, m2.w * alpha);
  atomicAdd(dp + 12, m3.x * alpha);
  atomicAdd(dp + 13, m3.y * alpha);
  atomicAdd(dp + 14, m3.z * alpha);
  atomicAdd(dp + 15, m3.w * alpha);
}

// ---------------- bias + ELU (in place, float4) -----------------------------
__global__ void bias_elu(float4* __restrict__ v, const float4* __restrict__ b) {
  const int i = blockIdx.x * blockDim.x + threadIdx.x;   // float4 index
  if (i >= N_NODES * (DIM / 4)) return;
  const float4 bb = b[i & (DIM / 4 - 1)];
  float4 x = v[i];
  x.x += bb.x; x.y += bb.y; x.z += bb.z; x.w += bb.w;
  x.x = x.x > 0.f ? x.x : expm1f(x.x);
  x.y = x.y > 0.f ? x.y : expm1f(x.y);
  x.z = x.z > 0.f ? x.z : expm1f(x.z);
  x.w = x.w > 0.f ? x.w : expm1f(x.w);
  v[i] = x;
}

// ---------------------------------------------------------------------------
static inline int cdiv(int a, int b) { return (a + b - 1) / b; }

extern "C" void kernel_launch(void* const* d_in, const int* in_sizes, int n_in,
                              void* d_out, int out_size, void* d_ws, size_t ws_size,
                              hipStream_t stream) {
  (void)in_sizes; (void)n_in; (void)out_size; (void)ws_size;
  const float*     x      = (const float*)d_in[0];
  const long long* ei     = (const long long*)d_in[1];   // int64 [2, E]
  const float*     W1     = (const float*)d_in[2];
  const float*     a1_src = (const float*)d_in[3];
  const float*     a1_dst = (const float*)d_in[4];
  const float*     b1     = (const float*)d_in[5];
  const float*     W2     = (const float*)d_in[6];
  const float*     a2_src = (const float*)d_in[7];
  const float*     a2_dst = (const float*)d_in[8];
  const float*     b2     = (const float*)d_in[9];
  const float*     Wp     = (const float*)d_in[10];
  const float*     bp     = (const float*)d_in[11];
  float*           out    = (float*)d_out;

  // workspace carve-out (~98 MB)
  float* ws      = (float*)d_ws;
  float* buf_xw  = ws;                                   // N*64
  float* buf_agg = buf_xw  + (size_t)N_NODES * DIM;      // N*64
  float* buf_als = buf_agg + (size_t)N_NODES * DIM;      // N*4
  float* buf_ald = buf_als + (size_t)N_NODES * HEADS;    // N*4
  float* buf_m   = buf_ald + (size_t)N_NODES * HEADS;    // N*4
  float* buf_s   = buf_m   + (size_t)N_NODES * HEADS;    // N*4
  float* buf_e   = buf_s   + (size_t)N_NODES * HEADS;    // E_TOT*4
  int2*  buf_sd  = (int2*)(buf_e + (size_t)E_TOT * HEADS); // E_TOT int2

  const int TB = 256;
  const int g_nh   = cdiv(N_NODES * HEADS, TB);
  const int g_nd4  = cdiv(N_NODES * (DIM / 4), TB);
  const int g_e    = cdiv(E_TOT, TB);
  const int g_eh   = cdiv(E_TOT * HEADS, TB);
  const int g_gemm = cdiv((N_NODES / 16) * 32, TB);      // one wave / 16-row strip

  build_edges<<<g_e, TB, 0, stream>>>(ei, buf_sd);

  // ---- layer 1 ----
  gemm_wmma_strip<<<g_gemm, TB, 0, stream>>>(x, W1, nullptr, buf_xw, N_NODES, 128);
  att_logits<<<g_nh, TB, 0, stream>>>(buf_xw, a1_src, a1_dst, buf_als, buf_ald);
  init_layer<<<g_nd4, TB, 0, stream>>>((float4*)buf_m, (float4*)buf_s, (float4*)buf_agg);
  edge_logits_max<<<g_e, TB, 0, stream>>>(buf_sd, (const float4*)buf_als,
                                          (const float4*)buf_ald, (float4*)buf_e, buf_m);
  edge_expsum<<<g_e, TB, 0, stream>>>(buf_sd, (const float4*)buf_m, (float4*)buf_e, buf_s);
  edge_scatter<<<g_eh, TB, 0, stream>>>(buf_sd, buf_e, buf_s, buf_xw, buf_agg);
  bias_elu<<<g_nd4, TB, 0, stream>>>((float4*)buf_agg, (const float4*)b1);

  // ---- layer 2 ----
  gemm_wmma_strip<<<g_gemm, TB, 0, stream>>>(buf_agg, W2, nullptr, buf_xw, N_NODES, DIM);
  att_logits<<<g_nh, TB, 0, stream>>>(buf_xw, a2_src, a2_dst, buf_als, buf_ald);
  init_layer<<<g_nd4, TB, 0, stream>>>((float4*)buf_m, (float4*)buf_s, (float4*)buf_agg);
  edge_logits_max<<<g_e, TB, 0, stream>>>(buf_sd, (const float4*)buf_als,
                                          (const float4*)buf_ald, (float4*)buf_e, buf_m);
  edge_expsum<<<g_e, TB, 0, stream>>>(buf_sd, (const float4*)buf_m, (float4*)buf_e, buf_s);
  edge_scatter<<<g_eh, TB, 0, stream>>>(buf_sd, buf_e, buf_s, buf_xw, buf_agg);
  bias_elu<<<g_nd4, TB, 0, stream>>>((float4*)buf_agg, (const float4*)b2);

  // ---- final projection: out = elu(h2) @ Wp + bp (WMMA) ----
  gemm_wmma_strip<<<g_gemm, TB, 0, stream>>>(buf_agg, Wp, bp, out, N_NODES, DIM);
}